// WaveCell_88141318848590
// MI455X (gfx1250) — compile-verified
//
#include <hip/hip_runtime.h>

// Problem geometry (fixed by the reference).
#define NZc 2048
#define NXc 2048
#define Bc  4
#define TZ  8                    // rows computed per workgroup
#define LDSW (NXc + 8)           // row stride in LDS: 4 pad floats each side (16B aligned)
#define NTHREADS 256

// ---- CDNA5 async global->LDS copy (ASYNCcnt path), with safe fallback ----
typedef int v4i __attribute__((vector_size(16)));
typedef __attribute__((address_space(1))) v4i* as1_v4i_ptr;
typedef __attribute__((address_space(3))) v4i* as3_v4i_ptr;

__device__ __forceinline__ void async_copy16(const float* __restrict__ g, float* l) {
#if __has_builtin(__builtin_amdgcn_global_load_async_to_lds_b128)
    // AS1 pointer value == generic value; AS3 pointer value == low 32 bits of
    // the generic LDS address (ISA 10.2).
    __builtin_amdgcn_global_load_async_to_lds_b128(
        (as1_v4i_ptr)(uintptr_t)g,
        (as3_v4i_ptr)(uintptr_t)l,
        0, 0);
#else
    *(float4*)l = *(const float4*)g;   // sync fallback: vmem load + ds store
#endif
}

__device__ __forceinline__ void wait_async_all() {
#if __has_builtin(__builtin_amdgcn_s_wait_asynccnt)
    __builtin_amdgcn_s_wait_asynccnt(0);
#else
    asm volatile("s_wait_asynccnt 0" ::: "memory");
#endif
}

__global__ void __launch_bounds__(NTHREADS)
wave_step_kernel(const float* __restrict__ h1,
                 const float* __restrict__ h2,
                 const float* __restrict__ cl,
                 const float* __restrict__ bd,
                 const float* __restrict__ dtp,
                 const float* __restrict__ hp,
                 float* __restrict__ yout,
                 float* __restrict__ h1out)
{
    extern __shared__ float lds[];            // (TZ+2) x LDSW floats (~82 KB)
    const int tid = threadIdx.x;
    const int z0  = blockIdx.x * TZ;          // first computed row
    const int bb  = blockIdx.y;               // batch
    const size_t base = (size_t)bb * NZc * NXc;

    // 1) Zero the Dirichlet pads (disjoint from async-load targets -> no race).
    for (int i = tid; i < (TZ + 2) * 4; i += NTHREADS) {
        int r = i >> 2, k = i & 3;
        lds[r * LDSW + k]            = 0.0f;  // left pad (cols 0..3)
        lds[r * LDSW + 4 + NXc + k]  = 0.0f;  // right pad
    }
    if (z0 == 0)
        for (int i = tid; i < NXc; i += NTHREADS) lds[4 + i] = 0.0f;               // top halo row
    if (z0 + TZ == NZc)
        for (int i = tid; i < NXc; i += NTHREADS) lds[(TZ + 1) * LDSW + 4 + i] = 0.0f; // bottom halo

    // 2) Async DMA-stage h1 rows z0-1 .. z0+TZ into LDS; prefetch pointwise streams.
    for (int i = tid; i < (TZ + 2) * (NXc / 4); i += NTHREADS) {
        const int r  = i >> 9;                // i / 512
        const int q  = i & 511;               // i % 512
        const int gz = z0 - 1 + r;
        if (0 <= gz && gz < NZc) {
            const size_t goff = base + (size_t)gz * NXc + 4 * q;
            async_copy16(h1 + goff, &lds[r * LDSW + 4 + 4 * q]);
            if ((q & 15) == 0 && gz >= z0 && gz < z0 + TZ) {
                const size_t poff = (size_t)gz * NXc + 4 * q;
                __builtin_prefetch(h2 + goff, 0, 0);
                __builtin_prefetch(cl + poff, 0, 0);
                __builtin_prefetch(bd + poff, 0, 0);
            }
        }
    }
    wait_async_all();          // this wave's ASYNCcnt == 0
    __syncthreads();           // all waves' LDS writes visible

    // 3) Scalar parameters (uniform -> s_load).
    const float dt      = *dtp;
    const float h       = *hp;
    const float inv_dt  = 1.0f / dt;
    const float inv_dt2 = inv_dt * inv_dt;
    const float inv_h2  = 1.0f / (h * h);

    // 4) Compute: 1 float4 of output per iteration, fully coalesced.
    for (int i = tid; i < TZ * (NXc / 4); i += NTHREADS) {
        const int zr = i >> 9;                // row within strip
        const int q  = i & 511;
        const int x  = 4 * q;
        const int r  = zr + 1;                // LDS row (halo offset)

        const float4 c4  = *(const float4*)&lds[r * LDSW + 4 + x];
        const float4 u4  = *(const float4*)&lds[(r - 1) * LDSW + 4 + x];
        const float4 d4  = *(const float4*)&lds[(r + 1) * LDSW + 4 + x];
        const float  lft = lds[r * LDSW + 3 + x];
        const float  rgt = lds[r * LDSW + 8 + x];

        float4 lap;
        lap.x = (u4.x + d4.x + lft  + c4.y - 4.0f * c4.x) * inv_h2;
        lap.y = (u4.y + d4.y + c4.x + c4.z - 4.0f * c4.y) * inv_h2;
        lap.z = (u4.z + d4.z + c4.y + c4.w - 4.0f * c4.z) * inv_h2;
        lap.w = (u4.w + d4.w + c4.z + rgt  - 4.0f * c4.w) * inv_h2;

        const size_t gi = base + (size_t)(z0 + zr) * NXc + x;
        const size_t pi =        (size_t)(z0 + zr) * NXc + x;
        const float4 h2v = *(const float4*)&h2[gi];
        const float4 cv  = *(const float4*)&cl[pi];
        const float4 bv  = *(const float4*)&bd[pi];

        float4 y;
        {
            const float den = 1.0f / (inv_dt2 + bv.x * inv_dt);
            y.x = den * (2.0f * inv_dt2 * c4.x - (inv_dt2 - bv.x * inv_dt) * h2v.x + cv.x * cv.x * lap.x);
        }
        {
            const float den = 1.0f / (inv_dt2 + bv.y * inv_dt);
            y.y = den * (2.0f * inv_dt2 * c4.y - (inv_dt2 - bv.y * inv_dt) * h2v.y + cv.y * cv.y * lap.y);
        }
        {
            const float den = 1.0f / (inv_dt2 + bv.z * inv_dt);
            y.z = den * (2.0f * inv_dt2 * c4.z - (inv_dt2 - bv.z * inv_dt) * h2v.z + cv.z * cv.z * lap.z);
        }
        {
            const float den = 1.0f / (inv_dt2 + bv.w * inv_dt);
            y.w = den * (2.0f * inv_dt2 * c4.w - (inv_dt2 - bv.w * inv_dt) * h2v.w + cv.w * cv.w * lap.w);
        }

        *(float4*)&yout[gi]  = y;   // output 0: y
        *(float4*)&h1out[gi] = c4;  // output 1: h1 passthrough (from LDS, no extra global read)
    }
}

extern "C" void kernel_launch(void* const* d_in, const int* in_sizes, int n_in,
                              void* d_out, int out_size, void* d_ws, size_t ws_size,
                              hipStream_t stream) {
    // setup_inputs order: h1, h2, c_linear, rho(unused), b, dt, h, t(unused), it(unused)
    const float* h1  = (const float*)d_in[0];
    const float* h2  = (const float*)d_in[1];
    const float* cl  = (const float*)d_in[2];
    const float* bd  = (const float*)d_in[4];
    const float* dtp = (const float*)d_in[5];
    const float* hp  = (const float*)d_in[6];

    float* y   = (float*)d_out;
    float* h1o = y + (size_t)Bc * NZc * NXc;

    dim3 grid(NZc / TZ, Bc);
    const size_t shmem = (size_t)(TZ + 2) * LDSW * sizeof(float);
    wave_step_kernel<<<grid, NTHREADS, shmem, stream>>>(h1, h2, cl, bd, dtp, hp, y, h1o);
}